// RelationalMultiheadAttention_36421322670630
// MI455X (gfx1250) — compile-verified
//
#include <hip/hip_runtime.h>
#include <hip/hip_bf16.h>

typedef float v2f __attribute__((ext_vector_type(2)));
typedef float v8f __attribute__((ext_vector_type(8)));
typedef unsigned int uint32x4 __attribute__((ext_vector_type(4)));
typedef int int32x8 __attribute__((ext_vector_type(8)));
typedef int int32x4 __attribute__((ext_vector_type(4)));

#define BB 2
#define QQ 2048
#define KK 2048
#define HH 8
#define DD 64
#define DVV 64
#define OUTN 512
#define EE 65536
#define BQ (BB*QQ)
#define KROW 68          // 64 + 4 DWORD TDM pad -> bank-conflict-free LDS rows
#define PROW 18          // padded P-transpose row stride (even, 8B-aligned b64)

__device__ __forceinline__ v8f wmma_f32(v2f a, v2f b, v8f c) {
    // V_WMMA_F32_16X16X4_F32 : D = A(16x4) x B(4x16) + C(16x16), all fp32
    return __builtin_amdgcn_wmma_f32_16x16x4_f32(
        false, a, false, b, (short)0, c, false, false);
}

// TDM: DMA a 2D fp32 tile (tile_d1 rows x 64 elements, row stride stride_d0
// elements) from global memory into LDS at byte offset lds_off, inserting
// 4 DWORDs of padding after every 64 DWORDs stored (pad_interval=5 -> 64 DW,
// pad_amount=3 -> 4 DW) so LDS rows are 68 floats: 16 lanes with row-stride
// access hit banks {0,4,...,60} -> conflict-free.
__device__ __forceinline__ void tdm_load_2d(unsigned lds_off, const float* g,
                                            int tile_d0, int tile_d1, int stride_d0) {
    unsigned long long ga = (unsigned long long)(uintptr_t)g;
    uint32x4 g0;
    g0.x = 1u;                                               // count=1, user mode
    g0.y = lds_off;                                          // lds_addr (bytes)
    g0.z = (unsigned)ga;                                     // global_addr[31:0]
    g0.w = (unsigned)((ga >> 32) & 0x01FFFFFFu) | (2u << 30);// [56:32] + type=2
    int32x8 g1;
    g1[0] = (int)((2u << 16)                                 // data_size = 4 bytes
                | (1u << 20)                                 // pad_enable
                | (5u << 22)                                 // pad_interval: 64 DW
                | (3u << 25));                               // pad_amount: 4 DW
    g1[1] = (tile_d0 & 0xFFFF) << 16;                        // tensor_dim0[15:0]
    g1[2] = (tile_d1 & 0xFFFF) << 16;                        // tensor_dim1[15:0]
    g1[3] = (tile_d0 & 0xFFFF) << 16;                        // tile_dim0
    g1[4] = (tile_d1 & 0xFFFF);                              // tile_dim1 (tile_dim2=0)
    g1[5] = stride_d0;                                       // tensor_dim0_stride[31:0]
    g1[6] = 0;
    g1[7] = 0;
    int32x4 z4 = {0, 0, 0, 0};                               // 2D: groups 2/3 unused
    int32x8 z8 = {0, 0, 0, 0, 0, 0, 0, 0};
    __builtin_amdgcn_tensor_load_to_lds(g0, g1, z4, z4, z8, 0);
}

// ---------------- K1: per-edge score bias  eb[e,h] = sum_d emb[type,h,d]*K[b,tgt,h,d]
__global__ void edge_bias_kernel(const float* __restrict__ Kp,
                                 const float* __restrict__ emb,
                                 const int* __restrict__ edges,
                                 const int* __restrict__ etypes,
                                 float* __restrict__ eb) {
    int wid  = (blockIdx.x * blockDim.x + threadIdx.x) >> 5;
    int lane = threadIdx.x & 31;
    if (wid >= EE) return;
    int sid = edges[wid*3+0];
    int tgt = edges[wid*3+2];
    int t   = etypes[wid];
    int h = lane >> 2, j = lane & 3;
    const float* kr = Kp + (((size_t)(sid*KK + tgt))*HH + h)*DD;
    const float* er = emb + (size_t)t*(HH*DD) + h*DD;
    float sum = 0.f;
    for (int d = j; d < DD; d += 4) sum += kr[d]*er[d];
    sum += __shfl_xor(sum, 1, 32);
    sum += __shfl_xor(sum, 2, 32);
    if (j == 0) eb[wid*HH + h] = sum;
}

// ---------------- K2: CSR of edges keyed by row = b*Q + q
__global__ void zero_kernel(int* __restrict__ p, int n) {
    int i = blockIdx.x * blockDim.x + threadIdx.x;
    if (i < n) p[i] = 0;
}
__global__ void count_kernel(const int* __restrict__ edges, int* __restrict__ cnt) {
    int e = blockIdx.x * blockDim.x + threadIdx.x;
    if (e < EE) atomicAdd(&cnt[edges[e*3+0]*QQ + edges[e*3+1]], 1);
}
__global__ void scan_kernel(const int* __restrict__ cnt,
                            int* __restrict__ rowStart, int* __restrict__ cursor) {
    if (threadIdx.x == 0 && blockIdx.x == 0) {
        int acc = 0;
        for (int i = 0; i < BQ; ++i) { rowStart[i] = acc; cursor[i] = acc; acc += cnt[i]; }
        rowStart[BQ] = acc;
    }
}
__global__ void fill_kernel(const int* __restrict__ edges,
                            int* __restrict__ cursor, int* __restrict__ lists) {
    int e = blockIdx.x * blockDim.x + threadIdx.x;
    if (e < EE) {
        int idx = atomicAdd(&cursor[edges[e*3+0]*QQ + edges[e*3+1]], 1);
        lists[idx] = e;
    }
}
__global__ void sort_kernel(const int* __restrict__ edges,
                            const int* __restrict__ rowStart, int* __restrict__ lists) {
    int row = blockIdx.x * blockDim.x + threadIdx.x;
    if (row >= BQ) return;
    int s = rowStart[row], e = rowStart[row+1];
    for (int i = s+1; i < e; ++i) {
        int ei = lists[i];
        long long key = (((long long)edges[ei*3+2]) << 20) | ei;
        int j = i - 1;
        while (j >= s) {
            int ej = lists[j];
            long long kj = (((long long)edges[ej*3+2]) << 20) | ej;
            if (kj <= key) break;
            lists[j+1] = ej; --j;
        }
        lists[j+1] = ei;
    }
}

// ---------------- K3: fused flash attention, TDM-staged K/V, fp32 WMMA
__global__ void __launch_bounds__(128)
attn_kernel(const float* __restrict__ Qp, const float* __restrict__ Kp,
            const float* __restrict__ Vp, const unsigned char* __restrict__ maskp,
            const int* __restrict__ rowStart, const int* __restrict__ lists,
            const int* __restrict__ edges, const float* __restrict__ eb,
            float* __restrict__ O, float* __restrict__ Mst, float* __restrict__ Lst) {
    __shared__ float ldsK[2][64*KROW];    // double-buffered, TDM-padded K tile
    __shared__ float ldsV[2][64*KROW];    // double-buffered, TDM-padded V tile
    __shared__ float ldsP[4][16*PROW];    // per-wave P transpose scratch (padded)
    const int lane = threadIdx.x & 31;
    const int wave = threadIdx.x >> 5;
    const int hl = lane >> 4;             // layout half
    const int m  = lane & 15;
    int bid = blockIdx.x;                 // B*H*(Q/64)
    int qt  = bid & 31;
    int h   = (bid >> 5) & 7;
    int b   = bid >> 8;
    int q0  = qt*64 + wave*16;

    // A-layout Q tile: lane m holds row q0+m, K-dim = 4*dc + 2*hl + {0,1}
    v2f qreg[16];
    const float* Qrow = Qp + (((size_t)(b*QQ + q0 + m))*HH + h)*DD;
    #pragma unroll
    for (int dc = 0; dc < 16; ++dc)
        qreg[dc] = *(const v2f*)(Qrow + dc*4 + hl*2);

    int ptr[16], pend[16];
    #pragma unroll
    for (int r = 0; r < 16; ++r) {
        int row = b*QQ + q0 + r;
        ptr[r]  = rowStart[row];
        pend[r] = rowStart[row+1];
    }

    v8f zero8 = {0,0,0,0,0,0,0,0};
    v8f oacc[4];
    #pragma unroll
    for (int c = 0; c < 4; ++c) oacc[c] = zero8;
    float mrow[8], lrow[8];
    #pragma unroll
    for (int r = 0; r < 8; ++r) { mrow[r] = -1e30f; lrow[r] = 0.f; }

    float* wl = &ldsP[wave][0];
    const int NG = KK/64;

    // prologue: TDM-load group 0 into buffer 0
    if (wave == 0) {
        tdm_load_2d((unsigned)(uintptr_t)&ldsK[0][0],
                    Kp + (((size_t)(b*KK + 0))*HH + h)*DD, DD, 64, HH*DD);
        tdm_load_2d((unsigned)(uintptr_t)&ldsV[0][0],
                    Vp + (((size_t)(b*KK + 0))*HH + h)*DVV, DVV, 64, HH*DVV);
    }

    for (int g = 0; g < NG; ++g) {
        int k0 = g*64;
        int cur = g & 1;
        __syncthreads();   // everyone done reading buffer (g+1)&1 from iteration g-1
        if (wave == 0) {
            if (g + 1 < NG) {
                int kn = (g+1)*64, nxt = (g+1) & 1;
                tdm_load_2d((unsigned)(uintptr_t)&ldsK[nxt][0],
                            Kp + (((size_t)(b*KK + kn))*HH + h)*DD, DD, 64, HH*DD);
                tdm_load_2d((unsigned)(uintptr_t)&ldsV[nxt][0],
                            Vp + (((size_t)(b*KK + kn))*HH + h)*DVV, DVV, 64, HH*DVV);
                __builtin_amdgcn_s_wait_tensorcnt(2);   // current group's 2 loads done
            } else {
                __builtin_amdgcn_s_wait_tensorcnt(0);
            }
        }
        __syncthreads();   // group g's K/V visible in LDS

        const float* Kt = &ldsK[cur][0];
        const float* Vt = &ldsV[cur][0];

        // S sub-tiles (64 columns = 4 x 16): 64 fp32 WMMAs
        v8f s0 = zero8, s1 = zero8, s2 = zero8, s3 = zero8;
        #pragma unroll
        for (int dc = 0; dc < 16; ++dc) {
            v2f a  = qreg[dc];
            int dof = dc*4 + hl*2;
            v2f b0 = *(const v2f*)(Kt + ( 0 + m)*KROW + dof);
            v2f b1 = *(const v2f*)(Kt + (16 + m)*KROW + dof);
            v2f b2 = *(const v2f*)(Kt + (32 + m)*KROW + dof);
            v2f b3 = *(const v2f*)(Kt + (48 + m)*KROW + dof);
            s0 = wmma_f32(a, b0, s0);
            s1 = wmma_f32(a, b1, s1);
            s2 = wmma_f32(a, b2, s2);
            s3 = wmma_f32(a, b3, s3);
        }

        // edge biases for tgt in [k0, k0+64)
        #pragma unroll
        for (int r = 0; r < 16; ++r) {
            while (ptr[r] < pend[r]) {
                int e2 = lists[ptr[r]];
                int tg = edges[e2*3+2];
                if (tg >= k0 + 64) break;
                float bv = eb[e2*HH + h];
                int rel = tg - k0;
                int t = rel >> 4, n = rel & 15;
                bool hit = (lane == (n + ((r >> 3) << 4)));
                if (hit && t == 0) s0[r & 7] += bv;
                if (hit && t == 1) s1[r & 7] += bv;
                if (hit && t == 2) s2[r & 7] += bv;
                if (hit && t == 3) s3[r & 7] += bv;
                ptr[r]++;
            }
        }

        // column mask (lane owns column k0 + t*16 + m)
        {
            const unsigned char* mp = maskp + b*KK + k0;
            float f0 = mp[ 0 + m] ? 0.f : 1.f;
            float a0 = mp[ 0 + m] ? -1e30f : 0.f;
            float f1 = mp[16 + m] ? 0.f : 1.f;
            float a1 = mp[16 + m] ? -1e30f : 0.f;
            float f2 = mp[32 + m] ? 0.f : 1.f;
            float a2 = mp[32 + m] ? -1e30f : 0.f;
            float f3 = mp[48 + m] ? 0.f : 1.f;
            float a3 = mp[48 + m] ? -1e30f : 0.f;
            #pragma unroll
            for (int i = 0; i < 8; ++i) {
                s0[i] = s0[i]*f0 + a0;
                s1[i] = s1[i]*f1 + a1;
                s2[i] = s2[i]*f2 + a2;
                s3[i] = s3[i]*f3 + a3;
            }
        }

        // online softmax over the 64-wide group (one reduction chain per row)
        #pragma unroll
        for (int r = 0; r < 8; ++r) {
            float v = fmaxf(fmaxf(s0[r], s1[r]), fmaxf(s2[r], s3[r]));
            v = fmaxf(v, __shfl_xor(v, 1, 32));
            v = fmaxf(v, __shfl_xor(v, 2, 32));
            v = fmaxf(v, __shfl_xor(v, 4, 32));
            v = fmaxf(v, __shfl_xor(v, 8, 32));
            float nm = fmaxf(mrow[r], v);
            float sc = __expf(mrow[r] - nm);
            float p0 = __expf(s0[r] - nm);
            float p1 = __expf(s1[r] - nm);
            float p2 = __expf(s2[r] - nm);
            float p3 = __expf(s3[r] - nm);
            s0[r] = p0; s1[r] = p1; s2[r] = p2; s3[r] = p3;
            float ps = (p0 + p1) + (p2 + p3);
            ps += __shfl_xor(ps, 1, 32);
            ps += __shfl_xor(ps, 2, 32);
            ps += __shfl_xor(ps, 4, 32);
            ps += __shfl_xor(ps, 8, 32);
            lrow[r] = lrow[r]*sc + ps;
            mrow[r] = nm;
            #pragma unroll
            for (int c = 0; c < 4; ++c) oacc[c][r] *= sc;
        }

        // O += P x V per 16-wide sub-tile (C->A transpose via per-wave LDS)
        #pragma unroll
        for (int t = 0; t < 4; ++t) {
            v8f st = (t == 0) ? s0 : (t == 1) ? s1 : (t == 2) ? s2 : s3;
            #pragma unroll
            for (int r = 0; r < 8; ++r)
                wl[(8*hl + r)*PROW + m] = st[r];
            #pragma unroll
            for (int kk = 0; kk < 4; ++kk) {
                v2f a = *(const v2f*)(wl + m*PROW + kk*4 + hl*2);
                const float* Vr0 = Vt + (t*16 + kk*4 + hl*2)*KROW;
                const float* Vr1 = Vr0 + KROW;
                #pragma unroll
                for (int c = 0; c < 4; ++c) {
                    v2f bv;
                    bv.x = Vr0[c*16 + m];
                    bv.y = Vr1[c*16 + m];
                    oacc[c] = wmma_f32(a, bv, oacc[c]);
                }
            }
        }
    }

    #pragma unroll
    for (int r = 0; r < 8; ++r) {
        int M = 8*hl + r;
        float inv = 1.0f / lrow[r];
        float* Orow = O + (((size_t)(b*QQ + q0 + M))*HH + h)*DVV;
        #pragma unroll
        for (int c = 0; c < 4; ++c) Orow[c*16 + m] = oacc[c][r] * inv;
        if (m == 0) {
            Mst[(size_t)(b*QQ + q0 + M)*HH + h] = mrow[r];
            Lst[(size_t)(b*QQ + q0 + M)*HH + h] = lrow[r];
        }
    }
}

// ---------------- K4: out[b,src] += p_e * value_bias  (one wave per (e,h))
__global__ void scatter_vbias(const float* __restrict__ Qp, const float* __restrict__ Kp,
                              const float* __restrict__ vbe,
                              const unsigned char* __restrict__ maskp,
                              const int* __restrict__ edges, const int* __restrict__ etypes,
                              const int* __restrict__ rowStart, const int* __restrict__ lists,
                              const float* __restrict__ eb,
                              const float* __restrict__ Mst, const float* __restrict__ Lst,
                              float* __restrict__ O) {
    int wid  = (blockIdx.x * blockDim.x + threadIdx.x) >> 5;
    int lane = threadIdx.x & 31;
    int e = wid >> 3, h = wid & 7;
    if (e >= EE) return;
    int sid = edges[e*3+0], src = edges[e*3+1], tgt = edges[e*3+2], t = etypes[e];
    if (maskp[sid*KK + tgt]) return;
    const float* qr = Qp + (((size_t)(sid*QQ + src))*HH + h)*DD;
    const float* kr = Kp + (((size_t)(sid*KK + tgt))*HH + h)*DD;
    float dot = qr[lane]*kr[lane] + qr[lane+32]*kr[lane+32];
    dot += __shfl_xor(dot, 16, 32);
    dot += __shfl_xor(dot, 8, 32);
    dot += __shfl_xor(dot, 4, 32);
    dot += __shfl_xor(dot, 2, 32);
    dot += __shfl_xor(dot, 1, 32);
    int row = sid*QQ + src;
    float bias = 0.f;
    for (int idx = rowStart[row]; idx < rowStart[row+1]; ++idx) {
        int e2 = lists[idx];
        if (edges[e2*3+2] == tgt) bias += eb[e2*HH + h];
    }
    float p = __expf(dot + bias - Mst[(size_t)row*HH + h]) / Lst[(size_t)row*HH + h];
    float* orow = O + ((size_t)row*HH + h)*DVV;
    const float* vr = vbe + (size_t)t*(HH*DVV) + h*DVV;
    atomicAdd(&orow[lane],      p*vr[lane]);
    atomicAdd(&orow[lane + 32], p*vr[lane + 32]);
}

// ---------------- K5: (4096 x 512) @ (512 x 512) fp32 WMMA projection
__global__ void __launch_bounds__(128)
out_gemm(const float* __restrict__ O, const float* __restrict__ W,
         float* __restrict__ Out) {
    const int lane = threadIdx.x & 31;
    const int wave = threadIdx.x >> 5;
    const int hl = lane >> 4, m = lane & 15;
    int bid = blockIdx.x;
    int nt = bid & 7, qt = bid >> 3;
    int q0 = qt*16;
    int n0 = nt*64 + wave*16;
    v8f acc = {0,0,0,0,0,0,0,0};
    const float* Arow = O + (size_t)(q0 + m)*(HH*DVV);
    for (int kk = 0; kk < (HH*DVV)/4; ++kk) {
        v2f a = *(const v2f*)(Arow + kk*4 + hl*2);
        v2f bv;
        bv.x = W[(size_t)(kk*4 + hl*2    )*OUTN + n0 + m];
        bv.y = W[(size_t)(kk*4 + hl*2 + 1)*OUTN + n0 + m];
        acc = wmma_f32(a, bv, acc);
    }
    #pragma unroll
    for (int r = 0; r < 8; ++r) {
        int M = 8*hl + r;
        Out[(size_t)(q0 + M)*OUTN + n0 + m] = acc[r];
    }
}

extern "C" void kernel_launch(void* const* d_in, const int* in_sizes, int n_in,
                              void* d_out, int out_size, void* d_ws, size_t ws_size,
                              hipStream_t stream) {
    const float* Qp   = (const float*)d_in[0];
    const float* Kp   = (const float*)d_in[1];
    const float* Vp   = (const float*)d_in[2];
    const float* ebe  = (const float*)d_in[3];
    const float* vbe  = (const float*)d_in[4];
    const float* Wout = (const float*)d_in[5];
    const unsigned char* maskp = (const unsigned char*)d_in[6];
    const int* edges  = (const int*)d_in[7];
    const int* etypes = (const int*)d_in[8];

    char* ws = (char*)d_ws;
    size_t off = 0;
    float* eb  = (float*)(ws + off); off += (size_t)EE*HH*4;
    float* Mst = (float*)(ws + off); off += (size_t)BQ*HH*4;
    float* Lst = (float*)(ws + off); off += (size_t)BQ*HH*4;
    float* O   = (float*)(ws + off); off += (size_t)BQ*HH*DVV*4;
    int* cnt      = (int*)(ws + off); off += (size_t)BQ*4;
    int* rowStart = (int*)(ws + off); off += (size_t)(BQ+1)*4 + 12;
    int* cursor   = (int*)(ws + off); off += (size_t)BQ*4;
    int* lists    = (int*)(ws + off); off += (size_t)EE*4;

    edge_bias_kernel<<<EE/8, 256, 0, stream>>>(Kp, ebe, edges, etypes, eb);
    zero_kernel<<<(BQ+255)/256, 256, 0, stream>>>(cnt, BQ);
    count_kernel<<<EE/256, 256, 0, stream>>>(edges, cnt);
    scan_kernel<<<1, 32, 0, stream>>>(cnt, rowStart, cursor);
    fill_kernel<<<EE/256, 256, 0, stream>>>(edges, cursor, lists);
    sort_kernel<<<(BQ+255)/256, 256, 0, stream>>>(edges, rowStart, lists);
    attn_kernel<<<BB*HH*(QQ/64), 128, 0, stream>>>(Qp, Kp, Vp, maskp, rowStart,
                                                   lists, edges, eb, O, Mst, Lst);
    scatter_vbias<<<(EE*HH)/8, 256, 0, stream>>>(Qp, Kp, vbe, maskp, edges, etypes,
                                                 rowStart, lists, eb, Mst, Lst, O);
    out_gemm<<<(BQ/16)*(OUTN/64), 128, 0, stream>>>(O, Wout, (float*)d_out);
}